// SharedTverskyInterpretable_5944234738211
// MI455X (gfx1250) — compile-verified
//
#include <hip/hip_runtime.h>

// SharedTverskyInterpretable for MI455X (gfx1250).
// out[b,p] = inter/(inter + a*xd + b*pd + eps) + bias[p], where
//   inter = 0.5*(Sx+Sp-A), xd = 0.5*(Sx-Sp+A), pd = 0.5*(Sp-Sx+A),
//   A[b,p] = sum_f |sig(x@W^T)[b,f] - sig(P)[p,f]|   (exact min/relu identity)
// GEMM runs on V_WMMA_F32_16X16X4_F32 (full f32, no precision downgrade).

#define B_DIM   2048
#define IN_DIM  512
#define NF_DIM  256
#define NP_DIM  256
#define EPSV    1e-8f

#define TM       16               // batch rows per block (one WMMA M-tile band)
#define PBLK     128              // prototypes per block
#define PSPLIT   (NP_DIM / PBLK)  // 2
#define PADF     (NF_DIM + 4)     // 260: conflict-free strides for b32/b128 LDS reads
#define PADK     (IN_DIM + 4)     // 516: conflict-free ds_load_b64 A-fragments
#define NTHREADS 256

typedef __attribute__((ext_vector_type(2))) float v2f;
typedef __attribute__((ext_vector_type(4))) float v4f;
typedef __attribute__((ext_vector_type(8))) float v8f;

__device__ __forceinline__ float sigmoidf(float v) {
    return 1.0f / (1.0f + __expf(-v));
}

__global__ __launch_bounds__(NTHREADS, 1)
void tversky_fused_kernel(const float* __restrict__ x,
                          const float* __restrict__ feat,    // (NF, IN) row-major
                          const float* __restrict__ proto,   // (NP, NF) row-major
                          const float* __restrict__ bias,    // (NP)
                          const float* __restrict__ alpha_p,
                          const float* __restrict__ beta_p,
                          float* __restrict__ out) {         // (B, NP)
    extern __shared__ float smem[];
    float* ldsP  = smem;                   // [PBLK][PADF] sigmoid(prototypes) half
    float* ldsA  = smem + PBLK * PADF;     // [TM][PADK]   x band (GEMM A staging)
    float* ldsXS = ldsA;                   // overlay after GEMM: [TM][PADF] x_sig

    const int tid  = threadIdx.x;
    const int wave = tid >> 5;
    const int lane = tid & 31;
    const int m0 = blockIdx.x * TM;
    const int p0 = blockIdx.y * PBLK;

    // ---- Stage 0: cooperative, coalesced LDS fills ----------------------
    for (int i = tid; i < TM * IN_DIM; i += NTHREADS) {
        const int r = i >> 9, c = i & (IN_DIM - 1);
        ldsA[r * PADK + c] = x[(m0 + r) * IN_DIM + c];
    }
    for (int i = tid; i < PBLK * NF_DIM; i += NTHREADS) {
        const int p = i >> 8, f = i & (NF_DIM - 1);
        ldsP[p * PADF + f] = sigmoidf(proto[(p0 + p) * NF_DIM + f]);
    }
    __syncthreads();

    // ---- Stage 1: x_feat tile via WMMA f32 16x16x4 -----------------------
    // A frag (16x4 f32): lanes 0-15 rows M, VGPR0/1 = K0/K1; lanes 16-31 = K2/K3.
    // B frag (4x16 f32): same striping with N on lanes (features^T[k][n] = feat[n][k]).
    const int mrow = lane & 15;
    const int koff = (lane >> 4) << 1;
    const int n0a = wave * 16;          // N-tile 0 for this wave
    const int n1a = (wave + 8) * 16;    // N-tile 1 (8 waves x 2 tiles = 256 feats)
    v8f accA = {};
    v8f accB = {};
    const float* arow  = &ldsA[mrow * PADK + koff];
    const float* brow0 = &feat[(n0a + mrow) * IN_DIM + koff];
    const float* brow1 = &feat[(n1a + mrow) * IN_DIM + koff];
#pragma unroll 4
    for (int k = 0; k < IN_DIM; k += 4) {
        const v2f a  = *(const v2f*)(arow + k);    // ds_load_2addr_b64, conflict-free
        const v2f b0 = *(const v2f*)(brow0 + k);   // global_load_b64, L2-resident
        const v2f b1 = *(const v2f*)(brow1 + k);
        accA = __builtin_amdgcn_wmma_f32_16x16x4_f32(
            false, a, false, b0, (short)0, accA, false, false);
        accB = __builtin_amdgcn_wmma_f32_16x16x4_f32(
            false, a, false, b1, (short)0, accB, false, false);
    }
    __syncthreads();   // everyone done reading ldsA before the overlay is written

    // Epilogue: sigmoid, scatter to overlay LDS (x_sig[b_local][f]).
    // C/D layout: VGPR v -> row v (lanes 0-15) / row v+8 (lanes 16-31), col = lane%16.
    const int rbase = (lane >> 4) << 3;
#pragma unroll
    for (int v = 0; v < 8; ++v) {
        const int row = v + rbase;
        ldsXS[row * PADF + n0a + mrow] = sigmoidf(accA[v]);
        ldsXS[row * PADF + n1a + mrow] = sigmoidf(accB[v]);
    }
    __syncthreads();

    // ---- Stage 2: Tversky via L1-distance identity -----------------------
    // 4 p-groups (32 lanes each) x 2 b-groups of 8 rows. Rows processed 4 at a
    // time so each prototype b128 load is amortized over 4 batch rows:
    // DS ops drop from 2/row/f4 to 1.25/row/f4 -> VALU-bound, dual-issue friendly.
    const float alpha = *alpha_p;
    const float beta  = *beta_p;
    const int pg = wave & 3;
    const int bg = wave >> 2;
    const int p     = pg * 32 + lane;
    const int pglob = p0 + p;
    const float bi = bias[pglob];
    const float* prow = &ldsP[p * PADF];

    float sp = 0.0f;                    // Sp[p] (per-lane)
    for (int f = 0; f < NF_DIM; f += 4) {
        const v4f pv = *(const v4f*)(prow + f);
        sp += (pv.x + pv.y) + (pv.z + pv.w);
    }

#pragma unroll
    for (int c = 0; c < 2; ++c) {
        const int bl0 = bg * 8 + c * 4;
        const float* xr0 = &ldsXS[(bl0 + 0) * PADF];
        const float* xr1 = &ldsXS[(bl0 + 1) * PADF];
        const float* xr2 = &ldsXS[(bl0 + 2) * PADF];
        const float* xr3 = &ldsXS[(bl0 + 3) * PADF];
        float l10 = 0.0f, l11 = 0.0f, l12 = 0.0f, l13 = 0.0f;
        float sx0 = 0.0f, sx1 = 0.0f, sx2 = 0.0f, sx3 = 0.0f;
#pragma unroll 2
        for (int f = 0; f < NF_DIM; f += 4) {
            const v4f pv = *(const v4f*)(prow + f);   // 1 load serves 4 rows
            const v4f a0 = *(const v4f*)(xr0 + f);    // uniform addr -> broadcast
            const v4f a1 = *(const v4f*)(xr1 + f);
            const v4f a2 = *(const v4f*)(xr2 + f);
            const v4f a3 = *(const v4f*)(xr3 + f);
            l10 += __builtin_fabsf(a0.x - pv.x) + __builtin_fabsf(a0.y - pv.y)
                 + __builtin_fabsf(a0.z - pv.z) + __builtin_fabsf(a0.w - pv.w);
            l11 += __builtin_fabsf(a1.x - pv.x) + __builtin_fabsf(a1.y - pv.y)
                 + __builtin_fabsf(a1.z - pv.z) + __builtin_fabsf(a1.w - pv.w);
            l12 += __builtin_fabsf(a2.x - pv.x) + __builtin_fabsf(a2.y - pv.y)
                 + __builtin_fabsf(a2.z - pv.z) + __builtin_fabsf(a2.w - pv.w);
            l13 += __builtin_fabsf(a3.x - pv.x) + __builtin_fabsf(a3.y - pv.y)
                 + __builtin_fabsf(a3.z - pv.z) + __builtin_fabsf(a3.w - pv.w);
            sx0 += (a0.x + a0.y) + (a0.z + a0.w);
            sx1 += (a1.x + a1.y) + (a1.z + a1.w);
            sx2 += (a2.x + a2.y) + (a2.z + a2.w);
            sx3 += (a3.x + a3.y) + (a3.z + a3.w);
        }
#pragma unroll
        for (int j = 0; j < 4; ++j) {
            const float sx = (j == 0) ? sx0 : (j == 1) ? sx1 : (j == 2) ? sx2 : sx3;
            const float l1 = (j == 0) ? l10 : (j == 1) ? l11 : (j == 2) ? l12 : l13;
            const float inter = 0.5f * (sx + sp - l1);
            const float xd    = 0.5f * (sx - sp + l1);
            const float pd    = 0.5f * (sp - sx + l1);
            const float denom = inter + alpha * xd + beta * pd + EPSV;
            out[(m0 + bl0 + j) * NP_DIM + pglob] = inter / denom + bi; // coalesced
        }
    }
}

extern "C" void kernel_launch(void* const* d_in, const int* in_sizes, int n_in,
                              void* d_out, int out_size, void* d_ws, size_t ws_size,
                              hipStream_t stream) {
    (void)in_sizes; (void)n_in; (void)out_size; (void)d_ws; (void)ws_size;
    const float* x     = (const float*)d_in[0];
    const float* feat  = (const float*)d_in[1];
    const float* proto = (const float*)d_in[2];
    const float* bias  = (const float*)d_in[3];
    const float* alpha = (const float*)d_in[4];
    const float* beta  = (const float*)d_in[5];
    float* out = (float*)d_out;

    const size_t smem_bytes = (size_t)(PBLK * PADF + TM * PADK) * sizeof(float); // ~166 KB
    // Raise dynamic-LDS cap (deterministic, not a stream op: graph-capture safe).
    (void)hipFuncSetAttribute(reinterpret_cast<const void*>(tversky_fused_kernel),
                              hipFuncAttributeMaxDynamicSharedMemorySize,
                              (int)smem_bytes);

    dim3 grid(B_DIM / TM, PSPLIT, 1);   // 128 x 2 = 256 workgroups
    tversky_fused_kernel<<<grid, NTHREADS, smem_bytes, stream>>>(
        x, feat, proto, bias, alpha, beta, out);
}